// RuleBasedDNF_20126216749736
// MI455X (gfx1250) — compile-verified
//
#include <hip/hip_runtime.h>
#include <cstdint>

// Problem constants from the reference:
// B=2048 batch, F=4096 features, C=8192 conjunctions, T=8 literals/conj,
// K=1024 classes, S=8 conjunctions/class. Output (B, K) float32.
#define BDIM 256
constexpr int Bc = 2048;
constexpr int Fc = 4096;
constexpr int Cc = 8192;
constexpr int Kc = 1024;
constexpr int ROWS = 2;                 // batch rows per workgroup

__device__ __forceinline__ void lit2(const float* __restrict__ sx0,
                                     const float* __restrict__ sx1,
                                     int f, int s, float& p0, float& p1) {
    float v0 = sx0[f];                  // ds_load_b32 gathers from LDS rows
    float v1 = sx1[f];
    p0 *= (s > 0) ? v0 : 1.0f - v0;
    p1 *= (s > 0) ? v1 : 1.0f - v1;
}

__global__ __launch_bounds__(BDIM) void rule_dnf_kernel(
    const float* __restrict__ x,          // (B, F)
    const int*   __restrict__ conj_feat,  // (C, 8)
    const int*   __restrict__ conj_sign,  // (C, 8)
    const int*   __restrict__ class_conj, // (K, 8)
    float*       __restrict__ out)        // (B, K)
{
    __shared__ float sx[ROWS][Fc];        // 32 KB: two x rows
    __shared__ float sconj[ROWS][Cc];     // 64 KB: conjunction values

    const int tid = threadIdx.x;
    const int b0  = blockIdx.x * ROWS;

    // Warm this block's first table lines into near caches (WGP scope) while
    // the async DMA below is in flight.
    __builtin_prefetch(conj_feat + (size_t)tid * 64, 0, 3);
    __builtin_prefetch(conj_sign + (size_t)tid * 64, 0, 3);
    __builtin_prefetch(class_conj + (size_t)tid * 8, 0, 3);

    // ---- Phase 1: async DMA x[b0:b0+2, :] (32 KB) from global into LDS -----
    // CDNA5 async global->LDS path: per-lane 16B chunks, tracked by ASYNCcnt.
    {
        const uint64_t gbase = (uint64_t)(uintptr_t)(x + (size_t)b0 * Fc);
        // Low 32 bits of a generic pointer into LDS == LDS byte address.
        const uint32_t lbase = (uint32_t)(uintptr_t)(&sx[0][0]);
        constexpr int NCHUNK = (ROWS * Fc * 4) / 16;   // 2048 x b128 chunks
        #pragma unroll
        for (int i = 0; i < NCHUNK / BDIM; ++i) {      // 8 chunks per thread
            const int      chunk = tid + i * BDIM;
            const uint32_t la    = lbase + (uint32_t)chunk * 16u;
            const uint64_t ga    = gbase + (uint64_t)chunk * 16u;
            asm volatile("global_load_async_to_lds_b128 %0, %1, off"
                         :: "v"(la), "v"(ga)
                         : "memory");
        }
        asm volatile("s_wait_asynccnt 0x0" ::: "memory");
    }
    __syncthreads();

    // ---- Phase 2: conjunction products for both rows (32 conjs/thread) -----
    {
        const int4* __restrict__ cf = (const int4*)conj_feat;
        const int4* __restrict__ cs = (const int4*)conj_sign;
        const float* __restrict__ sx0 = sx[0];
        const float* __restrict__ sx1 = sx[1];
        for (int c = tid; c < Cc; c += BDIM) {
            const int4 f0 = cf[2 * c + 0];
            const int4 f1 = cf[2 * c + 1];
            const int4 s0 = cs[2 * c + 0];
            const int4 s1 = cs[2 * c + 1];
            float p0 = 1.0f, p1 = 1.0f;
            lit2(sx0, sx1, f0.x, s0.x, p0, p1);
            lit2(sx0, sx1, f0.y, s0.y, p0, p1);
            lit2(sx0, sx1, f0.z, s0.z, p0, p1);
            lit2(sx0, sx1, f0.w, s0.w, p0, p1);
            lit2(sx0, sx1, f1.x, s1.x, p0, p1);
            lit2(sx0, sx1, f1.y, s1.y, p0, p1);
            lit2(sx0, sx1, f1.z, s1.z, p0, p1);
            lit2(sx0, sx1, f1.w, s1.w, p0, p1);
            sconj[0][c] = p0;
            sconj[1][c] = p1;
        }
    }
    __syncthreads();

    // ---- Phase 3: class max over 8 conjunctions, both rows (4 k/thread) ----
    {
        const int4* __restrict__ cc = (const int4*)class_conj;
        const float* __restrict__ c0 = sconj[0];
        const float* __restrict__ c1 = sconj[1];
        float* __restrict__ orow0 = out + (size_t)b0 * Kc;
        float* __restrict__ orow1 = orow0 + Kc;
        for (int k = tid; k < Kc; k += BDIM) {
            const int4 a = cc[2 * k + 0];
            const int4 d = cc[2 * k + 1];
            float m0 = c0[a.x], m1 = c1[a.x];
            m0 = fmaxf(m0, c0[a.y]);  m1 = fmaxf(m1, c1[a.y]);
            m0 = fmaxf(m0, c0[a.z]);  m1 = fmaxf(m1, c1[a.z]);
            m0 = fmaxf(m0, c0[a.w]);  m1 = fmaxf(m1, c1[a.w]);
            m0 = fmaxf(m0, c0[d.x]);  m1 = fmaxf(m1, c1[d.x]);
            m0 = fmaxf(m0, c0[d.y]);  m1 = fmaxf(m1, c1[d.y]);
            m0 = fmaxf(m0, c0[d.z]);  m1 = fmaxf(m1, c1[d.z]);
            m0 = fmaxf(m0, c0[d.w]);  m1 = fmaxf(m1, c1[d.w]);
            orow0[k] = m0;
            orow1[k] = m1;
        }
    }
}

extern "C" void kernel_launch(void* const* d_in, const int* in_sizes, int n_in,
                              void* d_out, int out_size, void* d_ws, size_t ws_size,
                              hipStream_t stream) {
    (void)in_sizes; (void)n_in; (void)out_size; (void)d_ws; (void)ws_size;
    const float* x          = (const float*)d_in[0];
    const int*   conj_feat  = (const int*)d_in[1];
    const int*   conj_sign  = (const int*)d_in[2];
    const int*   class_conj = (const int*)d_in[3];
    float*       out        = (float*)d_out;

    rule_dnf_kernel<<<Bc / ROWS, BDIM, 0, stream>>>(x, conj_feat, conj_sign,
                                                    class_conj, out);
}